// TileBasedGaussianRasterizer_90357521973627
// MI455X (gfx1250) — compile-verified
//
#include <hip/hip_runtime.h>
#include <hip/hip_bf16.h>

#define NG    512
#define HH    256
#define WW    256
#define TILE  16
#define TWX   16
#define TWY   16

typedef __attribute__((ext_vector_type(2))) float v2f;
typedef __attribute__((ext_vector_type(8))) float v8f;

// ---------------------------------------------------------------------------
// Kernel 1: per-gaussian preprocessing (projection, cov2d, conic, bounds, key)
// params layout (16 floats / gaussian):
//  0:mx 1:my 2:ia 3:ibc 4:id 5:opac 6:r 7:g 8:b 9:dep 10:txmin 11:txmax 12:tymin 13:tymax 14,15:pad
// ---------------------------------------------------------------------------
__global__ __launch_bounds__(256)
void gs_preprocess(const float* __restrict__ pos, const float* __restrict__ scl,
                   const float* __restrict__ rot, const float* __restrict__ col,
                   const float* __restrict__ opa, const float* __restrict__ Km,
                   float* __restrict__ params, unsigned long long* __restrict__ keys)
{
    int n = blockIdx.x * blockDim.x + threadIdx.x;
    if (n >= NG) return;

    const float fx = Km[0], cx = Km[2], fy = Km[4], cy = Km[5];
    const float X = pos[n*3+0], Y = pos[n*3+1], Z = pos[n*3+2];

    // quaternion -> rotation
    float qw = rot[n*4+0], qx = rot[n*4+1], qy = rot[n*4+2], qz = rot[n*4+3];
    float inv = 1.0f / sqrtf(qw*qw + qx*qx + qy*qy + qz*qz);
    qw *= inv; qx *= inv; qy *= inv; qz *= inv;
    float R[3][3];
    R[0][0] = 1.f - 2.f*(qy*qy + qz*qz); R[0][1] = 2.f*(qx*qy - qw*qz); R[0][2] = 2.f*(qx*qz + qw*qy);
    R[1][0] = 2.f*(qx*qy + qw*qz); R[1][1] = 1.f - 2.f*(qx*qx + qz*qz); R[1][2] = 2.f*(qy*qz - qw*qx);
    R[2][0] = 2.f*(qx*qz - qw*qy); R[2][1] = 2.f*(qy*qz + qw*qx); R[2][2] = 1.f - 2.f*(qx*qx + qy*qy);

    const float s0 = scl[n*3+0], s1 = scl[n*3+1], s2 = scl[n*3+2];
    float M[3][3];
    #pragma unroll
    for (int i = 0; i < 3; ++i) { M[i][0] = R[i][0]*s0; M[i][1] = R[i][1]*s1; M[i][2] = R[i][2]*s2; }
    float c3[3][3];
    #pragma unroll
    for (int i = 0; i < 3; ++i)
        #pragma unroll
        for (int k = 0; k < 3; ++k)
            c3[i][k] = M[i][0]*M[k][0] + M[i][1]*M[k][1] + M[i][2]*M[k][2];

    const float iz = 1.0f / Z;
    const float j00 = fx * iz, j02 = -fx * X * iz * iz;
    const float j11 = fy * iz, j12 = -fy * Y * iz * iz;
    // U = J * cov3 (2x3, exploiting J sparsity)
    float U00 = j00*c3[0][0] + j02*c3[2][0];
    float U01 = j00*c3[0][1] + j02*c3[2][1];
    float U02 = j00*c3[0][2] + j02*c3[2][2];
    float U10 = j11*c3[1][0] + j12*c3[2][0];
    float U11 = j11*c3[1][1] + j12*c3[2][1];
    float U12 = j11*c3[1][2] + j12*c3[2][2];
    const float EPS = 1e-4f;
    float a = U00*j00 + U02*j02 + EPS;
    float b = U01*j11 + U02*j12;
    float c = U10*j00 + U12*j02;
    float d = U11*j11 + U12*j12 + EPS;

    float mx = fx * X * iz + cx;
    float my = fy * Y * iz + cy;

    float trace = a + d;
    float det   = a*d - b*c;
    float disc  = fmaxf(trace*trace - 4.f*det, 0.f);
    float lam   = 0.5f * (trace + sqrtf(disc));
    float radius = 3.0f * sqrtf(lam + 1e-6f);

    float txmin = fminf(fmaxf(floorf((mx - radius) / (float)TILE), 0.f), (float)(TWX-1));
    float txmax = fminf(fmaxf(ceilf ((mx + radius) / (float)TILE), 0.f), (float)(TWX-1));
    float tymin = fminf(fmaxf(floorf((my - radius) / (float)TILE), 0.f), (float)(TWY-1));
    float tymax = fminf(fmaxf(ceilf ((my + radius) / (float)TILE), 0.f), (float)(TWY-1));

    float invdet = 1.0f / (a*d - b*c);
    float ia  =  d * invdet;
    float ibc = -(b + c) * invdet;
    float id_ =  a * invdet;

    float* p = params + n*16;
    p[0] = mx;  p[1] = my;  p[2] = ia;  p[3] = ibc; p[4] = id_;
    p[5] = opa[n];
    p[6] = col[n*3+0]; p[7] = col[n*3+1]; p[8] = col[n*3+2];
    p[9] = Z;
    p[10] = txmin; p[11] = txmax; p[12] = tymin; p[13] = tymax;
    p[14] = 0.f; p[15] = 0.f;

    // descending depth, stable (smaller original index first on ties)
    keys[n] = ((unsigned long long)__float_as_uint(Z) << 32)
            | (unsigned long long)(0xFFFFFFFFu - (unsigned)n);
}

// ---------------------------------------------------------------------------
// Kernel 2: bitonic sort (descending by key) + gather into sorted SoA tables.
// geomS: 8 floats/g  [mx,my,-ia/2,-ibc/2,-id/2,opac, packed_bounds(u32), pad]
//        (conic pre-scaled by -0.5 so the raster does w = __expf(m)*opac)
// attrS: 16 floats/g [r,g,b,dep,1, 0 x 11]  (one WMMA B-matrix row)
// ---------------------------------------------------------------------------
__global__ __launch_bounds__(256)
void gs_sort_gather(const float* __restrict__ params, const unsigned long long* __restrict__ keys,
                    float* __restrict__ geomS, float* __restrict__ attrS)
{
    __shared__ unsigned long long sk[NG];
    const int t = threadIdx.x;
    sk[t]       = keys[t];
    sk[t + 256] = keys[t + 256];

    for (int k = 2; k <= NG; k <<= 1) {
        for (int j = k >> 1; j > 0; j >>= 1) {
            __syncthreads();
            #pragma unroll
            for (int s = 0; s < 2; ++s) {
                int i   = t + (s << 8);
                int ixj = i ^ j;
                if (ixj > i) {
                    unsigned long long va = sk[i], vb = sk[ixj];
                    bool descSeg = ((i & k) == 0);
                    bool doSwap  = descSeg ? (va < vb) : (va > vb);
                    if (doSwap) { sk[i] = vb; sk[ixj] = va; }
                }
            }
        }
    }
    __syncthreads();

    #pragma unroll
    for (int s = 0; s < 2; ++s) {
        int i = t + (s << 8);
        unsigned idx = 0xFFFFFFFFu - (unsigned)(sk[i] & 0xFFFFFFFFull);
        const float* p = params + idx * 16;
        float* g = geomS + i * 8;
        g[0] = p[0]; g[1] = p[1];
        g[2] = -0.5f * p[2]; g[3] = -0.5f * p[3]; g[4] = -0.5f * p[4];
        g[5] = p[5];
        unsigned pb = ((unsigned)p[10]) | (((unsigned)p[11]) << 8)
                    | (((unsigned)p[12]) << 16) | (((unsigned)p[13]) << 24);
        g[6] = __uint_as_float(pb);
        g[7] = 0.f;
        float* a = attrS + i * 16;
        a[0] = p[6]; a[1] = p[7]; a[2] = p[8]; a[3] = p[9]; a[4] = 1.0f;
        #pragma unroll
        for (int q = 5; q < 16; ++q) a[q] = 0.f;
    }
}

// ---------------------------------------------------------------------------
// Kernel 3: rasterize. One block per 16x8 pixel strip; 8 waves, one row each.
// Visible gaussians are stream-compacted (order-preserving) once per block,
// then the dense loop runs ceil(nvis/4) chunks. Half-wave 0 evaluates
// gaussians k0,k1; half-wave 1 k2,k3 for the SAME 16 pixels, matching the
// 32-bit 16x4 A-matrix layout. D = pixels x {r,g,b,depth,alpha} accumulates
// via v_wmma_f32_16x16x4_f32.
// ---------------------------------------------------------------------------
__global__ __launch_bounds__(256)
void gs_raster(const float* __restrict__ geomS, const float* __restrict__ attrS,
               const float* __restrict__ bg, float* __restrict__ out)
{
    __shared__ float geom[(NG + 1) * 8];        // +1 dummy entry (index NG)
    __shared__ float attrs[(NG + 1) * 16];
    __shared__ unsigned short compact[NG + 4];  // compacted visible indices
    __shared__ unsigned segMask[16];            // per-32-gaussian vis ballots

    const int t   = threadIdx.x;
    const int tx  = blockIdx.x;       // tile x
    const int ty2 = blockIdx.y;       // half-tile y (8 rows)
    const int ty  = ty2 >> 1;         // tile y for culling
    const int lane = t & 31;
    const int wave = t >> 5;

    // --- cooperative stage of sorted tables into LDS (float4 vectorized) ---
    {
        const float4* g4 = (const float4*)geomS;
        float4* gl = (float4*)geom;
        #pragma unroll
        for (int i = 0; i < 4; ++i) gl[t + i * 256] = g4[t + i * 256];
        const float4* a4 = (const float4*)attrS;
        float4* al = (float4*)attrs;
        #pragma unroll
        for (int i = 0; i < 8; ++i) al[t + i * 256] = a4[t + i * 256];
    }
    if (t == 0) {   // dummy zero gaussian: w = exp(0)*0 = 0, attrs = 0
        #pragma unroll
        for (int q = 0; q < 8; ++q)  geom[NG * 8 + q] = 0.f;
        #pragma unroll
        for (int q = 0; q < 16; ++q) attrs[NG * 16 + q] = 0.f;
    }
    __syncthreads();

    // --- visibility + order-preserving stream compaction (ballot scan) ---
    bool vf0, vf1;
    {
        unsigned pb = __float_as_uint(geom[t * 8 + 6]);
        vf0 = ((int)(pb & 0xFF) <= tx) && (tx <= (int)((pb >> 8) & 0xFF)) &&
              ((int)((pb >> 16) & 0xFF) <= ty) && (ty <= (int)((pb >> 24) & 0xFF));
        unsigned pb2 = __float_as_uint(geom[(t + 256) * 8 + 6]);
        vf1 = ((int)(pb2 & 0xFF) <= tx) && (tx <= (int)((pb2 >> 8) & 0xFF)) &&
              ((int)((pb2 >> 16) & 0xFF) <= ty) && (ty <= (int)((pb2 >> 24) & 0xFF));
    }
    unsigned b0 = (unsigned)__ballot(vf0);
    unsigned b1 = (unsigned)__ballot(vf1);
    if (lane == 0) { segMask[wave] = b0; segMask[wave + 8] = b1; }
    __syncthreads();

    int off0 = 0, off1 = 0, total = 0;
    #pragma unroll
    for (int s = 0; s < 16; ++s) {
        if (s == wave)     off0 = total;
        if (s == wave + 8) off1 = total;
        total += __popc(segMask[s]);
    }
    const unsigned lmask = (1u << lane) - 1u;
    if (vf0) compact[off0 + __popc(b0 & lmask)] = (unsigned short)t;
    if (vf1) compact[off1 + __popc(b1 & lmask)] = (unsigned short)(t + 256);
    const int padded = (total + 3) & ~3;
    if (t < 4) {
        int p = total + t;
        if (p < padded) compact[p] = (unsigned short)NG;   // dummy
    }
    __syncthreads();

    // --- dense composite loop over visible gaussians only ---
    const int px   = lane & 15;       // pixel (M) and channel (N) index
    const int half = lane >> 4;
    const float xf = (float)(tx * TILE + px);
    const float yf = (float)(ty2 * 8 + wave);
    const int nchunk = padded >> 2;

    float T = 1.0f;
    v8f acc = {0.f, 0.f, 0.f, 0.f, 0.f, 0.f, 0.f, 0.f};

    for (int c = 0; c < nchunk; ++c) {
        unsigned pair = *(const unsigned*)(compact + c * 4 + 2 * half);
        const int idxA = (int)(pair & 0xFFFFu);   // K = 0/2 per half-wave
        const int idxB = (int)(pair >> 16);       // K = 1/3 per half-wave

        const float* pA = geom + idxA * 8;
        float dxA = xf - pA[0], dyA = yf - pA[1];
        float mA  = pA[2]*dxA*dxA + pA[3]*dxA*dyA + pA[4]*dyA*dyA;  // pre-scaled by -0.5
        float vA  = __expf(mA) * pA[5];
        float wA  = (vA > 0.01f) ? vA : 0.f;

        const float* pB = geom + idxB * 8;
        float dxB = xf - pB[0], dyB = yf - pB[1];
        float mB  = pB[2]*dxB*dxB + pB[3]*dxB*dyB + pB[4]*dyB*dyB;
        float vB  = __expf(mB) * pB[5];
        float wB  = (vB > 0.01f) ? vB : 0.f;

        // exclusive-prefix transmittance across the 4-gaussian chunk
        float pHalf  = (1.f - wA) * (1.f - wB);
        float pOther = __shfl_xor(pHalf, 16, 32);     // swap half-wave products
        float prefix = half ? (T * pOther) : T;       // lanes 16-31 see (1-w0)(1-w1)

        v2f A;
        A.x = prefix * wA;                    // contrib K0 / K2
        A.y = prefix * (1.f - wA) * wB;       // contrib K1 / K3
        T   = T * pHalf * pOther;

        v2f B;
        B.x = attrs[idxA * 16 + px];          // row K0/K2, column = channel
        B.y = attrs[idxB * 16 + px];          // row K1/K3

        // D[pixel 16 x channel 16] += A[16x4] * B[4x16]
        acc = __builtin_amdgcn_wmma_f32_16x16x4_f32(
                  false, A, false, B, (short)0, acc, false, false);
    }

    // --- scatter: lane holds channel (lane&15) for pixels M = r + 8*half ---
    float* img = out;
    float* dep = out + HH * WW * 3;
    float* alp = dep + HH * WW;
    const int ch = px;
    const int yI = ty2 * 8 + wave;
    const float bgc = (ch < 3) ? bg[ch] : 0.f;
    #pragma unroll
    for (int r = 0; r < 8; ++r) {
        int M  = r + 8 * half;
        int xI = tx * TILE + M;
        float v = acc[r];
        if (ch < 3) {
            float o = fminf(fmaxf(bgc + v, 0.f), 1.f);
            img[(yI * WW + xI) * 3 + ch] = o;
        } else if (ch == 3) {
            dep[yI * WW + xI] = v;
        } else if (ch == 4) {
            alp[yI * WW + xI] = v;
        }
    }
}

// ---------------------------------------------------------------------------
extern "C" void kernel_launch(void* const* d_in, const int* in_sizes, int n_in,
                              void* d_out, int out_size, void* d_ws, size_t ws_size,
                              hipStream_t stream) {
    (void)in_sizes; (void)n_in; (void)out_size; (void)ws_size;
    const float* pos = (const float*)d_in[0];
    const float* scl = (const float*)d_in[1];
    const float* rot = (const float*)d_in[2];
    const float* col = (const float*)d_in[3];
    const float* opa = (const float*)d_in[4];
    const float* Km  = (const float*)d_in[5];
    const float* bg  = (const float*)d_in[6];
    float* out = (float*)d_out;

    char* ws = (char*)d_ws;
    float*              params = (float*)(ws + 0);                    // 32 KB
    unsigned long long* keys   = (unsigned long long*)(ws + 32768);   //  4 KB
    float*              geomS  = (float*)(ws + 36864);                // 16 KB
    float*              attrS  = (float*)(ws + 53248);                // 32 KB

    gs_preprocess<<<dim3(2), dim3(256), 0, stream>>>(pos, scl, rot, col, opa, Km, params, keys);
    gs_sort_gather<<<dim3(1), dim3(256), 0, stream>>>(params, keys, geomS, attrS);
    gs_raster<<<dim3(TWX, TWY * 2), dim3(256), 0, stream>>>(geomS, attrS, bg, out);
}